// PanopticSegmentationGeneratorV2_16080357556723
// MI455X (gfx1250) — compile-verified
//
#include <hip/hip_runtime.h>
#include <stdint.h>

// ---------------------------------------------------------------------------
// PanopticSegmentationGeneratorV2 for gfx1250 (MI455X, wave32)
//   B=2, N=100 detections, masks 28x28, canvas 512x512, seg 128x128x32.
// Outputs (concatenated int32): instance_mask[B*H*W], category_mask[B*H*W].
// ---------------------------------------------------------------------------

typedef float v2f __attribute__((ext_vector_type(2)));
typedef float v8f __attribute__((ext_vector_type(8)));

#define BB   2
#define NN   100
#define MH   28
#define MW   28
#define HH   512
#define WW   512
#define HWP  (HH * WW)          // 262144 pixels per batch
#define NPIX (BB * HWP)         // 524288
#define SC   32                 // seg channels
#define SH   128                // seg input spatial

__device__ __forceinline__ int clampi(int v, int lo, int hi) {
    return v < lo ? lo : (v > hi ? hi : v);
}

// ---------------------------------------------------------------------------
// 0) init: instance_mask = -1 (VOID_INSTANCE), category accumulator = 0,
//    per-class stuff counts = 0.
// ---------------------------------------------------------------------------
__global__ void init_kernel(int* __restrict__ out_inst, int* __restrict__ out_cat,
                            int* __restrict__ counts, int n) {
    int i = blockIdx.x * blockDim.x + threadIdx.x;
    if (i < n) { out_inst[i] = -1; out_cat[i] = 0; }
    if (i < BB * SC) counts[i] = 0;
}

// ---------------------------------------------------------------------------
// 1) Segmentation path: bilinear 128->512 upsample as two chained 16x16 f32
//    WMMA matmuls per (tile, channel), argmax over channels in D-layout regs,
//    per-class stuff-area counts via LDS + one global atomic per class.
//    One wave per 16x16 output tile; 8 waves (256 thr) per block; 256 blocks.
// ---------------------------------------------------------------------------
__global__ __launch_bounds__(256) void seg_resize_argmax_kernel(
    const float* __restrict__ seg, uint8_t* __restrict__ segcls,
    int* __restrict__ counts) {
    const int t    = threadIdx.x;
    const int lane = t & 31;
    const int wv   = t >> 5;
    const int tile = blockIdx.x * 8 + wv;   // 0..2047 (1024 tiles per batch)
    const int b    = tile >> 10;
    const int tt   = tile & 1023;
    const int ty   = tt >> 5;
    const int tx   = tt & 31;
    // 16-row/col source windows; always cover the (clamped) 6 needed taps.
    const int by = clampi(ty * 4 - 1, 0, SH - 16);
    const int bx = clampi(tx * 4 - 1, 0, SH - 16);

    __shared__ float xch[8][256];   // per-wave D->A relayout buffer
    __shared__ int   cnt[SC];
    if (t < SC) cnt[t] = 0;
    __syncthreads();

    const int ml = lane & 15;   // M (A) or N (B,C,D) index for this lane
    const int hi = lane >> 4;   // K-half select per 16x4/4x16 layout

    // A_y: weight of src row (by+k) for output row ty*16+ml  (half-pixel centers)
    float ay[8];
    {
        int   y  = ty * 16 + ml;
        float sy = (float)y * 0.25f - 0.375f;
        float fy = floorf(sy);
        float ly = sy - fy;
        int   y0 = clampi((int)fy, 0, SH - 1);
        int   y1 = clampi((int)fy + 1, 0, SH - 1);
#pragma unroll
        for (int s = 0; s < 4; ++s)
#pragma unroll
            for (int e = 0; e < 2; ++e) {
                int   row = by + 4 * s + 2 * hi + e;
                float w   = 0.f;
                if (row == y0) w += 1.f - ly;
                if (row == y1) w += ly;
                ay[s * 2 + e] = w;
            }
    }
    // B2 = A_x^T: weight of src col (bx+k) for output col tx*16+ml
    float bw[8];
    {
        int   xo = tx * 16 + ml;
        float sx = (float)xo * 0.25f - 0.375f;
        float fx = floorf(sx);
        float lx = sx - fx;
        int   x0 = clampi((int)fx, 0, SH - 1);
        int   x1 = clampi((int)fx + 1, 0, SH - 1);
#pragma unroll
        for (int s = 0; s < 4; ++s)
#pragma unroll
            for (int e = 0; e < 2; ++e) {
                int   col = bx + 4 * s + 2 * hi + e;
                float w   = 0.f;
                if (col == x0) w += 1.f - lx;
                if (col == x1) w += lx;
                bw[s * 2 + e] = w;
            }
    }

    float bestv[8];
    int   besti[8];
#pragma unroll
    for (int r = 0; r < 8; ++r) { bestv[r] = -3.0e38f; besti[r] = 0; }

    const float* base = seg + (size_t)b * SH * SH * SC;

    for (int c = 0; c < SC; ++c) {
        // Stage 1: T(16 outY x 16 srcX) = A_y(16x16) * S_patch(16x16)
        v8f acc = {0.f, 0.f, 0.f, 0.f, 0.f, 0.f, 0.f, 0.f};
#pragma unroll
        for (int s = 0; s < 4; ++s) {
            v2f a;
            a.x = ay[2 * s];
            a.y = ay[2 * s + 1];
            const int k0 = 4 * s + 2 * hi;
            v2f bm;
            bm.x = base[(size_t)((by + k0) * SH + bx + ml) * SC + c];
            bm.y = base[(size_t)((by + k0 + 1) * SH + bx + ml) * SC + c];
            acc = __builtin_amdgcn_wmma_f32_16x16x4_f32(false, a, false, bm,
                                                        (short)0, acc, false, false);
        }
        // D-layout -> LDS (row-major 16x16), then re-read in A-layout.
#pragma unroll
        for (int r = 0; r < 8; ++r) xch[wv][(r + 8 * hi) * 16 + ml] = acc[r];
        __syncthreads();
        // Stage 2: Out(16x16) = T(16x16) * A_x^T(16x16)
        v8f o2 = {0.f, 0.f, 0.f, 0.f, 0.f, 0.f, 0.f, 0.f};
#pragma unroll
        for (int s = 0; s < 4; ++s) {
            const int k0 = 4 * s + 2 * hi;
            v2f a2;
            a2.x = xch[wv][ml * 16 + k0];
            a2.y = xch[wv][ml * 16 + k0 + 1];
            v2f b2;
            b2.x = bw[2 * s];
            b2.y = bw[2 * s + 1];
            o2 = __builtin_amdgcn_wmma_f32_16x16x4_f32(false, a2, false, b2,
                                                       (short)0, o2, false, false);
        }
        __syncthreads();
        // Running argmax (first max wins -> strict >)
#pragma unroll
        for (int r = 0; r < 8; ++r) {
            float v  = o2[r];
            bool  gt = v > bestv[r];
            bestv[r] = gt ? v : bestv[r];
            besti[r] = gt ? c : besti[r];
        }
    }

    // Write class map + stuff-area counts (stuff: cls != 0 (void) && != 1 (things))
    const int xo = tx * 16 + ml;
#pragma unroll
    for (int r = 0; r < 8; ++r) {
        int y   = ty * 16 + r + 8 * hi;
        int p   = (b * HH + y) * WW + xo;
        int cls = besti[r];
        segcls[p] = (uint8_t)cls;
        if (cls > 1) atomicAdd(&cnt[cls], 1);
    }
    __syncthreads();
    const int bblk = (blockIdx.x * 8) >> 10;   // batch is uniform per block
    if (t < SC && cnt[t] != 0) atomicAdd(&counts[bblk * SC + t], cnt[t]);
}

// ---------------------------------------------------------------------------
// 2) Instance resolution: one 1024-thread workgroup per batch image.
//    Claim state = 32KB LDS bitmask. 100-step scan in score order: paste the
//    28x28 mask on the fly (mask cached in LDS), area/overlap by reduction,
//    accept if overlap/area < 0.5, claim free pixels, write inst/cat.
// ---------------------------------------------------------------------------
__global__ __launch_bounds__(1024) void resolve_kernel(
    const float* __restrict__ boxes, const int* __restrict__ klass,
    const float* __restrict__ scores, const float* __restrict__ masks,
    int* __restrict__ out_inst, int* __restrict__ out_cat) {
    const int b    = blockIdx.x;
    const int t    = threadIdx.x;
    const int lane = t & 31;
    const int wv   = t >> 5;

    __shared__ unsigned int claimed[HWP / 32];  // 8192 words = 32KB
    __shared__ float        msk[MH * MW];
    __shared__ int          order[NN];
    __shared__ int          s_area, s_over, s_accept;

    for (int i = t; i < HWP / 32; i += 1024) claimed[i] = 0;
    if (t < NN) {  // stable rank of -score (ties -> lower index first)
        float si = scores[b * NN + t];
        int   r  = 0;
        for (int j = 0; j < NN; ++j) {
            float sj = scores[b * NN + j];
            r += (sj > si) || (sj == si && j < t);
        }
        order[r] = t;
    }
    __syncthreads();

    const int   x    = t & 511;    // this thread's fixed column
    const int   yoff = t >> 9;     // 0 or 1; rows y = 2u + yoff
    const float gx   = (float)x;

    for (int k = 0; k < NN; ++k) {
        const int   idx = order[k];
        const int   cls = klass[b * NN + idx];
        const float sc  = scores[b * NN + idx];
        if (!(sc > 0.5f && cls != 0)) continue;  // keep-mask zeroes -> no-op step

        const float* bx4  = boxes + (size_t)(b * NN + idx) * 4;
        const float  ymin = bx4[0], xmin = bx4[1], ymax = bx4[2], xmax = bx4[3];
        for (int i = t; i < MH * MW; i += 1024)
            msk[i] = masks[(size_t)(b * NN + idx) * (MH * MW) + i];
        if (t == 0) { s_area = 0; s_over = 0; }
        __syncthreads();

        const float hsy = (ymax > ymin) ? (float)MH / (ymax - ymin) : 0.f;
        const float wsx = (xmax > xmin) ? (float)MW / (xmax - xmin) : 0.f;
        const float mxf = (gx + 0.5f - xmin) * wsx - 0.5f;
        const float x0f = floorf(mxf);
        const float lx  = mxf - x0f;
        const int   x0  = clampi((int)x0f, 0, MW - 1);
        const int   x1  = clampi((int)x0f + 1, 0, MW - 1);
        const bool  xv  = (gx >= xmin) && (gx < xmax);

        unsigned int bmreg[8];
#pragma unroll
        for (int i = 0; i < 8; ++i) bmreg[i] = 0;
        int area = 0, over = 0;

        for (int u = 0; u < 256; ++u) {           // pixel p = u*1024 + t
            const int   y   = u * 2 + yoff;
            const float gy  = (float)y;
            const float myf = (gy + 0.5f - ymin) * hsy - 0.5f;
            const float y0f = floorf(myf);
            const float ly  = myf - y0f;
            const int   y0  = clampi((int)y0f, 0, MH - 1);
            const int   y1  = clampi((int)y0f + 1, 0, MH - 1);
            const float v00 = msk[y0 * MW + x0], v01 = msk[y0 * MW + x1];
            const float v10 = msk[y1 * MW + x0], v11 = msk[y1 * MW + x1];
            const float val = (1.f - ly) * ((1.f - lx) * v00 + lx * v01) +
                              ly * ((1.f - lx) * v10 + lx * v11);
            const bool bm = xv && (gy >= ymin) && (gy < ymax) && (val > 0.5f);
            const unsigned int word = claimed[u * 32 + wv];   // p>>5 == u*32+wv
            const bool         cl   = (word >> lane) & 1u;
            area += bm;
            over += (bm && cl);
            bmreg[u >> 5] |= ((unsigned int)bm) << (u & 31);
        }
        atomicAdd(&s_area, area);
        atomicAdd(&s_over, over);
        __syncthreads();
        if (t == 0) {
            float frac = (s_area == 0) ? 0.f
                                       : (float)s_over / fmaxf((float)s_area, 1.f);
            s_accept = (frac < 0.5f) ? 1 : 0;
        }
        __syncthreads();
        if (s_accept) {
            for (int u = 0; u < 256; ++u) {
                const bool bm = (bmreg[u >> 5] >> (u & 31)) & 1u;
                const unsigned int word = claimed[u * 32 + wv];
                const bool nb = bm && !((word >> lane) & 1u);
                const unsigned int m32 = (unsigned int)__ballot(nb);
                if (lane == 0 && m32) claimed[u * 32 + wv] = word | m32;
                if (nb) {
                    const int p = u * 1024 + t;
                    out_inst[b * HWP + p] = idx + 1;
                    out_cat[b * HWP + p]  = cls;
                }
            }
        }
        __syncthreads();
    }
}

// ---------------------------------------------------------------------------
// 3) Merge stuff + things into final category_mask (in place over out_cat).
// ---------------------------------------------------------------------------
__global__ void merge_kernel(const uint8_t* __restrict__ segcls,
                             const int* __restrict__ counts,
                             int* __restrict__ out_cat, int n) {
    const int p = blockIdx.x * blockDim.x + threadIdx.x;
    if (p >= n) return;
    const int   b    = p >> 18;  // / HWP
    const int   cls  = segcls[p];
    const float cf   = (float)counts[b * SC + cls];
    const bool  kept = (cls > 1) && (cf > 4096.0f);
    const int   stuff_cat = kept ? cls : 0;
    const int   inst_cat  = out_cat[p];
    out_cat[p] = (stuff_cat != 0 && inst_cat == 0) ? (stuff_cat + 90) : inst_cat;
}

// ---------------------------------------------------------------------------
extern "C" void kernel_launch(void* const* d_in, const int* in_sizes, int n_in,
                              void* d_out, int out_size, void* d_ws, size_t ws_size,
                              hipStream_t stream) {
    (void)in_sizes; (void)n_in; (void)out_size; (void)ws_size;
    const float* boxes  = (const float*)d_in[0];   // (2,100,4)   f32
    const int*   klass  = (const int*)d_in[1];     // (2,100)     i32
    const float* scores = (const float*)d_in[2];   // (2,100)     f32
    const float* masks  = (const float*)d_in[3];   // (2,100,28,28) f32
    const float* seg    = (const float*)d_in[4];   // (2,128,128,32) f32

    int* out_inst = (int*)d_out;            // instance_mask, NPIX int32
    int* out_cat  = out_inst + NPIX;        // category_mask, NPIX int32

    uint8_t* segcls = (uint8_t*)d_ws;                        // NPIX bytes
    int*     counts = (int*)((char*)d_ws + (size_t)NPIX);    // B*32 ints

    init_kernel<<<(NPIX + 255) / 256, 256, 0, stream>>>(out_inst, out_cat, counts, NPIX);
    seg_resize_argmax_kernel<<<256, 256, 0, stream>>>(seg, segcls, counts);
    resolve_kernel<<<BB, 1024, 0, stream>>>(boxes, klass, scores, masks, out_inst, out_cat);
    merge_kernel<<<(NPIX + 255) / 256, 256, 0, stream>>>(segcls, counts, out_cat, NPIX);
}